// ModelNew_80908593922602
// MI455X (gfx1250) — compile-verified
//
#include <hip/hip_runtime.h>
#include <math.h>

// Problem constants (from reference): B=16, IC=16, OC=64, K=3, H=W=256
#define BB 16
#define IC 16
#define OC 64
#define KK 3
#define HH 256
#define WW 256

// Tile: 32x8 output pixels per 256-thread block
#define TW 32
#define TH 8
#define TCOLS (TW + KK - 1)   // 34
#define TROWS (TH + KK - 1)   // 10
#define LSTRIDE 36            // padded row stride in LDS (conflict-free)
#define TAPS (IC * KK * KK)   // 144 taps per pixel
#define NPAIR (TAPS / 2)      // 72 tap pairs
#define FLT_BIG 3.402823466e38f

#if __has_builtin(__builtin_amdgcn_global_load_async_to_lds_b32)
#define USE_ASYNC_LDS 1
#else
#define USE_ASYNC_LDS 0
#endif

typedef __attribute__((address_space(1))) int g_int;  // global (device) memory
typedef __attribute__((address_space(3))) int l_int;  // LDS
typedef float v2f __attribute__((ext_vector_type(2)));

__device__ __forceinline__ float tanh_hw(float v) {
#if __has_builtin(__builtin_amdgcn_tanhf)
    return __builtin_amdgcn_tanhf(v);        // CDNA5 TRANS op: v_tanh_f32
#elif __has_builtin(__builtin_amdgcn_tanh_f32)
    return __builtin_amdgcn_tanh_f32(v);
#else
    return tanhf(v);
#endif
}

// Packed 2xf32 multiply: taps in a VGPR pair, uniform weights in an SGPR pair.
// VOP3P v_pk_mul_f32 with explicit zero OPSEL (required on gfx1250 for pk-f32):
// lo = a.lo * w.lo, hi = a.hi * w.hi  -> 2 candidates per VALU instruction.
__device__ __forceinline__ v2f pk_mul(v2f a, v2f w) {
#if defined(__gfx1250__)
    v2f d;
    asm("v_pk_mul_f32 %0, %1, %2 op_sel:[0,0] op_sel_hi:[0,0]"
        : "=v"(d) : "v"(a), "s"(w));
    return d;
#else
    return a * w;
#endif
}

// 3-input min in one VALU op: acc = min(acc, min(a, b))
__device__ __forceinline__ float min3_hw(float acc, float a, float b) {
#if defined(__gfx1250__)
    float d;
    asm("v_min3_num_f32 %0, %1, %2, %3" : "=v"(d) : "v"(acc), "v"(a), "v"(b));
    return d;
#else
    return fminf(acc, fminf(a, b));
#endif
}

__global__ __launch_bounds__(256) void fused_minconv_kernel(
    const float* __restrict__ x,     // [B, IC, H, W]
    const float* __restrict__ wgt,   // [OC, IC, 3, 3] -> contiguous 144 per oc
    const float* __restrict__ bias,  // [OC]
    float* __restrict__ out)         // [B, 1, H, W]
{
    __shared__ float xs[IC][TROWS][LSTRIDE];

    const int tid = threadIdx.x;
    const int b  = blockIdx.z;
    const int h0 = blockIdx.y * TH;
    const int w0 = blockIdx.x * TW;

    // ---- Stage 1: fill LDS tile (zero-padded on the high side) ----
    const float* xb = x + (size_t)b * IC * HH * WW;
    for (int e = tid; e < IC * TROWS * TCOLS; e += 256) {
        const int ic = e / (TROWS * TCOLS);
        const int rr = (e / TCOLS) % TROWS;
        const int cc = e % TCOLS;
        const int gh = h0 + rr;
        const int gw = w0 + cc;
        float* dst = &xs[ic][rr][cc];
        const bool inb = (gh < HH) & (gw < WW);
#if USE_ASYNC_LDS
        if (inb) {
            // CDNA5 async copy engine: global -> LDS, tracked by ASYNCcnt
            __builtin_amdgcn_global_load_async_to_lds_b32(
                (g_int*)(xb + ((size_t)ic * HH + gh) * WW + gw),
                (l_int*)dst,
                0, 0);
        } else {
            *dst = 0.0f;  // ds_store zero halo (disjoint addresses from async writes)
        }
#else
        *dst = inb ? xb[((size_t)ic * HH + gh) * WW + gw] : 0.0f;
#endif
    }

#if USE_ASYNC_LDS
#if __has_builtin(__builtin_amdgcn_s_wait_asynccnt)
    __builtin_amdgcn_s_wait_asynccnt(0);
#else
    asm volatile("s_wait_asynccnt 0" ::: "memory");
#endif
#endif
    __syncthreads();

    // ---- Stage 2: pull this pixel's 144 taps into a pair-aligned VGPR file ----
    const int px = tid & (TW - 1);   // lane -> column: conflict-free LDS reads
    const int py = tid >> 5;
    const int h = h0 + py;
    const int w = w0 + px;

    v2f xr2[NPAIR];                  // tap order == weight layout: t = ic*9 + kh*3 + kw
#pragma unroll
    for (int ic = 0; ic < IC; ++ic)
#pragma unroll
        for (int kh = 0; kh < KK; ++kh)
#pragma unroll
            for (int kw = 0; kw < KK; ++kw) {
                const int t = ic * 9 + kh * 3 + kw;
                xr2[t >> 1][t & 1] = xs[ic][py + kh][px + kw];
            }

    // ---- Stage 3: min over (oc, taps) ----
    // Forced v_pk_mul_f32 + forced v_min3_num_f32:
    //   2 VALU per 2 candidates = 1.0 op/candidate (op-count floor for this reduction).
    float best = FLT_BIG;
#pragma unroll 1
    for (int oc = 0; oc < OC; ++oc) {
        const v2f* __restrict__ wo2 = (const v2f*)(wgt + oc * TAPS);  // 8B-aligned (oc*576B)
        float m0 = FLT_BIG, m1 = FLT_BIG, m2 = FLT_BIG, m3 = FLT_BIG;
#pragma unroll
        for (int j = 0; j < NPAIR; j += 4) {
            const v2f p0 = pk_mul(xr2[j + 0], wo2[j + 0]);
            const v2f p1 = pk_mul(xr2[j + 1], wo2[j + 1]);
            const v2f p2 = pk_mul(xr2[j + 2], wo2[j + 2]);
            const v2f p3 = pk_mul(xr2[j + 3], wo2[j + 3]);
            m0 = min3_hw(m0, p0.x, p0.y);
            m1 = min3_hw(m1, p1.x, p1.y);
            m2 = min3_hw(m2, p2.x, p2.y);
            m3 = min3_hw(m3, p3.x, p3.y);
        }
        const float m = fminf(fminf(m0, m1), fminf(m2, m3));
        best = fminf(best, m + bias[oc]);  // bias folded once per oc (exact)
    }

    const float r = tanh_hw(tanh_hw(best));  // v_tanh_f32 x2 (TRANS pipe)
    out[(size_t)b * (HH * WW) + h * WW + w] = r;
}

extern "C" void kernel_launch(void* const* d_in, const int* in_sizes, int n_in,
                              void* d_out, int out_size, void* d_ws, size_t ws_size,
                              hipStream_t stream) {
    const float* x    = (const float*)d_in[0];
    const float* wgt  = (const float*)d_in[1];
    const float* bias = (const float*)d_in[2];
    float* out = (float*)d_out;

    dim3 grid(WW / TW, HH / TH, BB);  // (8, 32, 16) — exact tiling, no tails
    dim3 block(256, 1, 1);            // 8 waves (wave32) per block
    fused_minconv_kernel<<<grid, block, 0, stream>>>(x, wgt, bias, out);
}